// AttentionHead_78168404787269
// MI455X (gfx1250) — compile-verified
//
#include <hip/hip_runtime.h>
#include <hip/hip_bf16.h>
#include <math.h>

// ---------------- problem constants (match reference) ----------------
#define N_NODES 8192
#define IN_DIM  512
#define OUT_DIM 64
#define N_EDGES 262144
#define SLOPE   0.2f

typedef __attribute__((ext_vector_type(2))) float v2f;
typedef __attribute__((ext_vector_type(8))) float v8f;

// =====================================================================
// Kernel 1: Wh = X @ W^T + b  via V_WMMA_F32_16X16X4_F32
// One wave -> 16x64 output block. 512 waves total.
// A (16x4 f32): lane l<16 holds A[m0+l, k..k+1] in v[0..1];
//               lane l>=16 holds A[m0+(l-16), k+2..k+3]   (ISA 7.12.2)
// B (4x16 f32): symmetric K split; N = lane&15.
// C/D (16x16 f32, 8 VGPR): vgpr r -> M=r (lanes 0-15) / M=8+r (16-31), N=lane&15.
// =====================================================================
__global__ void __launch_bounds__(256)
gat_wh_gemm(const float* __restrict__ X,     // [N_NODES, IN_DIM]
            const float* __restrict__ Wl,    // [OUT_DIM, IN_DIM]
            const float* __restrict__ bl,    // [OUT_DIM]
            float* __restrict__ Wh)          // [N_NODES, OUT_DIM]
{
    const int lane = threadIdx.x & 31;
    const int wave = (blockIdx.x * blockDim.x + threadIdx.x) >> 5;
    const int m0   = wave * 16;
    if (m0 >= N_NODES) return;               // wave-uniform: EXEC stays all-1s

    const int mrow = m0 + (lane & 15);
    const int koff = (lane >> 4) * 2;        // 0 or 2
    const int ncol = lane & 15;

    const float* __restrict__ arow = X + (size_t)mrow * IN_DIM + koff;
    const float* __restrict__ b0   = Wl + (size_t)(ncol +  0) * IN_DIM + koff;
    const float* __restrict__ b1   = Wl + (size_t)(ncol + 16) * IN_DIM + koff;
    const float* __restrict__ b2   = Wl + (size_t)(ncol + 32) * IN_DIM + koff;
    const float* __restrict__ b3   = Wl + (size_t)(ncol + 48) * IN_DIM + koff;

    v8f acc0 = {}, acc1 = {}, acc2 = {}, acc3 = {};

#pragma unroll 4
    for (int k = 0; k < IN_DIM; k += 4) {
        v2f a  = *(const v2f*)(arow + k);
        v2f q0 = *(const v2f*)(b0 + k);
        v2f q1 = *(const v2f*)(b1 + k);
        v2f q2 = *(const v2f*)(b2 + k);
        v2f q3 = *(const v2f*)(b3 + k);
        acc0 = __builtin_amdgcn_wmma_f32_16x16x4_f32(false, a, false, q0, (short)0, acc0, false, false);
        acc1 = __builtin_amdgcn_wmma_f32_16x16x4_f32(false, a, false, q1, (short)0, acc1, false, false);
        acc2 = __builtin_amdgcn_wmma_f32_16x16x4_f32(false, a, false, q2, (short)0, acc2, false, false);
        acc3 = __builtin_amdgcn_wmma_f32_16x16x4_f32(false, a, false, q3, (short)0, acc3, false, false);
    }

    const int rsel = (lane >> 4) * 8;        // lanes 16-31 hold M = r+8
#pragma unroll
    for (int r = 0; r < 8; ++r) {
        const int row = m0 + r + rsel;
        float* __restrict__ o = Wh + (size_t)row * OUT_DIM;
        o[ 0 + ncol] = acc0[r] + bl[ 0 + ncol];
        o[16 + ncol] = acc1[r] + bl[16 + ncol];
        o[32 + ncol] = acc2[r] + bl[32 + ncol];
        o[48 + ncol] = acc3[r] + bl[48 + ncol];
    }
}

// =====================================================================
// Kernel 2: per-node attention partials aU[i]=Watt[0:64]·Wh[i],
// aV[i]=Watt[64:128]·Wh[i], plus column sums totalWh[d]=Σ_i Wh[i][d]
// (LDS-accumulated, one small batch of global atomics per block).
// =====================================================================
__global__ void __launch_bounds__(256)
gat_scores(const float* __restrict__ Wh,
           const float* __restrict__ Watt,   // [2*OUT_DIM]
           float* __restrict__ aU, float* __restrict__ aV,
           float* __restrict__ totalWh)      // [OUT_DIM], pre-zeroed
{
    __shared__ float ls[OUT_DIM];
    if (threadIdx.x < OUT_DIM) ls[threadIdx.x] = 0.0f;
    __syncthreads();

    const int lane  = threadIdx.x & 31;
    const int wib   = threadIdx.x >> 5;                 // wave in block (0..7)
    const int gwave = blockIdx.x * 8 + wib;
    const int nwave = gridDim.x * 8;

    const float wu0 = Watt[lane],      wu1 = Watt[lane + 32];
    const float wv0 = Watt[64 + lane], wv1 = Watt[96 + lane];

    float c0 = 0.0f, c1 = 0.0f;
    for (int i = gwave; i < N_NODES; i += nwave) {
        const float x0 = Wh[(size_t)i * OUT_DIM + lane];
        const float x1 = Wh[(size_t)i * OUT_DIM + lane + 32];
        c0 += x0; c1 += x1;
        float pu = x0 * wu0 + x1 * wu1;
        float pv = x0 * wv0 + x1 * wv1;
#pragma unroll
        for (int off = 16; off > 0; off >>= 1) {
            pu += __shfl_xor(pu, off, 32);
            pv += __shfl_xor(pv, off, 32);
        }
        if (lane == 0) { aU[i] = pu; aV[i] = pv; }
    }
    atomicAdd(&ls[lane], c0);
    atomicAdd(&ls[lane + 32], c1);
    __syncthreads();
    if (threadIdx.x < OUT_DIM) atomicAdd(&totalWh[threadIdx.x], ls[threadIdx.x]);
}

// =====================================================================
// CSR build: count -> scan -> fill (edge weights w = expm1(lrelu(e)))
// =====================================================================
__global__ void __launch_bounds__(256)
gat_edge_count(const int* __restrict__ U, int* __restrict__ cnt)
{
    const int k = blockIdx.x * blockDim.x + threadIdx.x;
    if (k < N_EDGES) atomicAdd(&cnt[U[k]], 1);
}

__global__ void __launch_bounds__(1024)
gat_scan(const int* __restrict__ cnt,
         int* __restrict__ rowStart,        // [N_NODES+1]
         int* __restrict__ cursor)          // [N_NODES]
{
    __shared__ int s[1024];
    const int t = threadIdx.x;
    const int base = t * 8;
    int v[8]; int sum = 0;
#pragma unroll
    for (int j = 0; j < 8; ++j) { v[j] = cnt[base + j]; sum += v[j]; }
    s[t] = sum;
    __syncthreads();
    // Hillis-Steele inclusive scan over 1024 partials
    for (int off = 1; off < 1024; off <<= 1) {
        int x = (t >= off) ? s[t - off] : 0;
        __syncthreads();
        s[t] += x;
        __syncthreads();
    }
    int run = s[t] - sum;                   // exclusive prefix for this thread
#pragma unroll
    for (int j = 0; j < 8; ++j) {
        rowStart[base + j] = run;
        cursor[base + j]   = run;
        run += v[j];
    }
    if (t == 1023) rowStart[N_NODES] = s[1023];
}

__global__ void __launch_bounds__(256)
gat_edge_fill(const int* __restrict__ U, const int* __restrict__ V,
              const float* __restrict__ aU, const float* __restrict__ aV,
              const float* __restrict__ bAtt,
              int* __restrict__ cursor,
              int* __restrict__ cols, float* __restrict__ wvals)
{
    const int k = blockIdx.x * blockDim.x + threadIdx.x;
    if (k >= N_EDGES) return;
    const int u = U[k], v = V[k];
    float e = aU[u] + aV[v] + bAtt[0];
    float l = (e > 0.0f) ? e : SLOPE * e;   // leaky_relu
    float w = expm1f(l);                    // exp(lrelu(e)) - 1
    const int pos = atomicAdd(&cursor[u], 1);
    cols[pos]  = v;
    wvals[pos] = w;
}

// =====================================================================
// Kernel 6: wave-per-row sparse gather + normalization + relu.
//   out[i] = relu((Σ_j w_ij·Wh[col]+totalWh) / (Σ_j w_ij + N))
// Wh is 2 MB -> L2 resident; each edge costs one coalesced 256B read.
// =====================================================================
__global__ void __launch_bounds__(256)
gat_row_aggregate(const int* __restrict__ rowStart,
                  const int* __restrict__ cols,
                  const float* __restrict__ wvals,
                  const float* __restrict__ Wh,
                  const float* __restrict__ totalWh,
                  float* __restrict__ out)
{
    const int lane = threadIdx.x & 31;
    const int row  = (blockIdx.x * blockDim.x + threadIdx.x) >> 5;
    if (row >= N_NODES) return;

    const int beg = rowStart[row], end = rowStart[row + 1];
    float a0 = 0.0f, a1 = 0.0f, z = 0.0f;
    for (int j = beg; j < end; ++j) {
        const float w = wvals[j];           // wave-uniform load
        const int   c = cols[j];
        const float* __restrict__ wr = Wh + (size_t)c * OUT_DIM;
        a0 += w * wr[lane];
        a1 += w * wr[lane + 32];
        z  += w;
    }
    const float inv = 1.0f / (z + (float)N_NODES);
    float o0 = (a0 + totalWh[lane])      * inv;
    float o1 = (a1 + totalWh[lane + 32]) * inv;
    out[(size_t)row * OUT_DIM + lane]      = o0 > 0.0f ? o0 : 0.0f;
    out[(size_t)row * OUT_DIM + lane + 32] = o1 > 0.0f ? o1 : 0.0f;
}

// =====================================================================
extern "C" void kernel_launch(void* const* d_in, const int* in_sizes, int n_in,
                              void* d_out, int out_size, void* d_ws, size_t ws_size,
                              hipStream_t stream)
{
    const float* X    = (const float*)d_in[0];   // [8192,512]
    const float* Wl   = (const float*)d_in[1];   // [64,512]
    const float* bl   = (const float*)d_in[2];   // [64]
    const float* Watt = (const float*)d_in[3];   // [128]
    const float* bAtt = (const float*)d_in[4];   // [1]
    const int*   U    = (const int*)d_in[5];     // [262144]
    const int*   V    = (const int*)d_in[6];     // [262144]
    float*       out  = (float*)d_out;           // [8192,64]

    // workspace carve-up (~4.4 MB)
    float* Wh       = (float*)d_ws;              // 8192*64
    float* aU       = Wh + (size_t)N_NODES * OUT_DIM;
    float* aV       = aU + N_NODES;
    float* totalWh  = aV + N_NODES;              // 64
    float* wvals    = totalWh + OUT_DIM;         // 262144
    int*   cnt      = (int*)(wvals + N_EDGES);   // 8192
    int*   rowStart = cnt + N_NODES;             // 8193
    int*   cursor   = rowStart + (N_NODES + 1);  // 8192
    int*   cols     = cursor + N_NODES;          // 262144

    hipMemsetAsync(cnt, 0, N_NODES * sizeof(int), stream);
    hipMemsetAsync(totalWh, 0, OUT_DIM * sizeof(float), stream);

    // 1) GEMM via f32 WMMA: 512 waves -> 64 blocks x 256
    gat_wh_gemm<<<N_NODES / (16 * 8), 256, 0, stream>>>(X, Wl, bl, Wh);
    // 2) per-node scores + column sums
    gat_scores<<<64, 256, 0, stream>>>(Wh, Watt, aU, aV, totalWh);
    // 3-5) CSR build
    gat_edge_count<<<N_EDGES / 256, 256, 0, stream>>>(U, cnt);
    gat_scan<<<1, 1024, 0, stream>>>(cnt, rowStart, cursor);
    gat_edge_fill<<<N_EDGES / 256, 256, 0, stream>>>(U, V, aU, aV, bAtt,
                                                     cursor, cols, wvals);
    // 6) wave-per-row gather + softmax-normalize + relu
    gat_row_aggregate<<<N_NODES * 32 / 256, 256, 0, stream>>>(rowStart, cols, wvals,
                                                              Wh, totalWh, out);
}